// GNNModel_19739669692575
// MI455X (gfx1250) — compile-verified
//
#include <hip/hip_runtime.h>
#include <hip/hip_bf16.h>
#include <stdint.h>
#include <stddef.h>

typedef __attribute__((ext_vector_type(2))) float v2f;
typedef __attribute__((ext_vector_type(8))) float v8f;

#define BB 8
#define NNODE 20000
#define EDGES 320000
#define DHID 128
#define BNROWS (BB * NNODE)   // 160000 rows, divisible by 16

// ---------------------------------------------------------------- utilities

__global__ void zero_f32v4(float4* __restrict__ p, long n4) {
    long i = (long)blockIdx.x * blockDim.x + threadIdx.x;
    long stride = (long)gridDim.x * blockDim.x;
    const float4 z = make_float4(0.f, 0.f, 0.f, 0.f);
    for (; i < n4; i += stride) p[i] = z;
}

__global__ void deg_kernel(const int* __restrict__ dst, float* __restrict__ deg) {
    int e = blockIdx.x * blockDim.x + threadIdx.x;
    if (e < EDGES) atomicAdd(&deg[dst[e]], 1.0f);
}

// Edge scatter-add. Each block handles G edges: stages their src/dst in LDS
// once (kills the B*d-fold redundant index loads), then each thread moves one
// float4 per step (coalesced b128 load + 4 f32 atomics that stay in L2 --
// x/h buffers are 41-82 MB << 192 MB L2).
template <int DSHIFT>   // d = 1<<DSHIFT (64 or 128)
__global__ __launch_bounds__(256) void scatter_add(const float* __restrict__ xin,
                                                   const int* __restrict__ src,
                                                   const int* __restrict__ dst,
                                                   float* __restrict__ agg) {
    constexpr int G   = 8;                       // edges per block
    constexpr int d   = 1 << DSHIFT;
    constexpr int Q   = (BB * d) >> 2;           // float4 units per edge (128/256)
    __shared__ int ssrc[G], sdst[G];

    const int tid  = threadIdx.x;
    const int ebase = blockIdx.x * G;
    if (tid < G) {
        ssrc[tid] = src[ebase + tid];
        sdst[tid] = dst[ebase + tid];
    }
    __syncthreads();

    // G*Q float4 units, blockDim=256 threads
    constexpr int ITERS = (G * Q) / 256;
#pragma unroll
    for (int it = 0; it < ITERS; ++it) {
        const int idx = it * 256 + tid;
        const int e   = idx / Q;                 // edge within group
        const int r4  = idx % Q;                 // float4 index within (B,d)
        const int b   = r4 >> (DSHIFT - 2);
        const int f4  = r4 & ((d >> 2) - 1);
        const float4 v = ((const float4*)(xin + ((size_t)b * NNODE + ssrc[e]) * d))[f4];
        float* ap = agg + ((size_t)b * NNODE + sdst[e]) * d + (f4 << 2);
        atomicAdd(ap + 0, v.x);
        atomicAdd(ap + 1, v.y);
        atomicAdd(ap + 2, v.z);
        atomicAdd(ap + 3, v.w);
    }
}

// ------------------------------------------------- WMMA node update (fused)
//   y = diag(1/deg) * (Wl @ agg) + Wr @ x + bl
// Weights staged in LDS once per block; inner loop = 8 batched ds_load_b64
// B-fragments then 8 back-to-back V_WMMA_F32_16X16X4_F32. Two accumulator
// banks (neighbor/self) so the 1/deg scale moves to the epilogue and no VALU
// op sits on the WMMA critical path.
template <int DIN, bool RELU_OUT, bool DUAL_STORE>
__global__ __launch_bounds__(256) void sage_node_update(
        const float* __restrict__ agg,
        const float* __restrict__ xin,
        const float* __restrict__ Wl,
        const float* __restrict__ bl,
        const float* __restrict__ Wr,
        const float* __restrict__ deg,
        float* __restrict__ out,    // gets ReLU if RELU_OUT
        float* __restrict__ out2,   // raw copy if DUAL_STORE
        int ntiles) {
    __shared__ float sW[2 * DHID * DIN];         // Wl then Wr (64KB / 128KB)

    const int tid  = threadIdx.x;
    const int wave = tid >> 5;
    const int lane = tid & 31;
    const int tile = blockIdx.x * (blockDim.x >> 5) + wave;
    const bool active = (tile < ntiles);         // wave-uniform

    const int m  = lane & 15;                    // A row / B,C,D column in tile
    const int kh = (lane >> 4) << 1;             // K sub-offset: 0 or 2

    const size_t row = (size_t)(active ? tile : 0) * 16 + m;
    const float* arow = agg + row * DIN;
    const float* xrow = xin + row * DIN;
    if (active) {                                // get A rows in flight early
        __builtin_prefetch(arow, 0, 0);
        __builtin_prefetch(xrow, 0, 0);
    }

    // cooperative weight stage: global -> LDS, float4-coalesced
    constexpr int NW4 = DHID * DIN / 4;
    for (int i = tid; i < NW4; i += 256) ((float4*)sW)[i]       = ((const float4*)Wl)[i];
    for (int i = tid; i < NW4; i += 256) ((float4*)sW)[NW4 + i] = ((const float4*)Wr)[i];
    __syncthreads();
    if (!active) return;

    const float* sWl = sW;
    const float* sWr = sW + DHID * DIN;

    v8f cl[8], cr[8];
#pragma unroll
    for (int t = 0; t < 8; ++t) { cl[t] = (v8f)0.0f; cr[t] = (v8f)0.0f; }

    // neighbor term: A = agg rows (unscaled), B = Wl^T fragments from LDS
    for (int kb = 0; kb < DIN; kb += 4) {
        const v2f a = *(const v2f*)(arow + kb + kh);
        v2f b[8];
#pragma unroll
        for (int t = 0; t < 8; ++t)
            b[t] = *(const v2f*)(sWl + (size_t)(t * 16 + m) * DIN + kb + kh);
#pragma unroll
        for (int t = 0; t < 8; ++t)
            cl[t] = __builtin_amdgcn_wmma_f32_16x16x4_f32(
                false, a, false, b[t], (short)0, cl[t], false, false);
    }
    // self term: A = x rows, B = Wr^T fragments from LDS
    for (int kb = 0; kb < DIN; kb += 4) {
        const v2f a = *(const v2f*)(xrow + kb + kh);
        v2f b[8];
#pragma unroll
        for (int t = 0; t < 8; ++t)
            b[t] = *(const v2f*)(sWr + (size_t)(t * 16 + m) * DIN + kb + kh);
#pragma unroll
        for (int t = 0; t < 8; ++t)
            cr[t] = __builtin_amdgcn_wmma_f32_16x16x4_f32(
                false, a, false, b[t], (short)0, cr[t], false, false);
    }

    // epilogue: per-OUTPUT-row 1/deg scale, bias, (relu), store.
    // D element (vgpr i, lane): row = i + 8*(lane>=16), col = m.
    const int mhi = (lane >> 4) << 3;            // 0 or 8
    float invd[8];
#pragma unroll
    for (int i = 0; i < 8; ++i) {
        const int orow = tile * 16 + mhi + i;
        invd[i] = 1.0f / fmaxf(deg[orow % NNODE], 1.0f);
    }
#pragma unroll
    for (int t = 0; t < 8; ++t) {
        const float bias = bl[t * 16 + m];
#pragma unroll
        for (int i = 0; i < 8; ++i) {
            const size_t orow = (size_t)tile * 16 + mhi + i;
            const float  v    = cl[t][i] * invd[i] + cr[t][i] + bias;
            if (DUAL_STORE) out2[orow * DHID + t * 16 + m] = v;
            out[orow * DHID + t * 16 + m] = RELU_OUT ? fmaxf(v, 0.0f) : v;
        }
    }
}

// ------------------------------------------- final layer (D_OUT = 3, VALU)
__global__ void sage_out_layer(const float* __restrict__ agg,
                               const float* __restrict__ r,
                               const float* __restrict__ Wld,
                               const float* __restrict__ bld,
                               const float* __restrict__ Wrd,
                               const float* __restrict__ deg,
                               float* __restrict__ out) {
    const int row = blockIdx.x * blockDim.x + threadIdx.x;
    if (row >= BNROWS) return;
    const int   node = row % NNODE;
    const float invd = 1.0f / fmaxf(deg[node], 1.0f);
    const float4* arow = (const float4*)(agg + (size_t)row * DHID);
    const float4* xrow = (const float4*)(r   + (size_t)row * DHID);
    float acc0 = 0.f, acc1 = 0.f, acc2 = 0.f;
    for (int k4 = 0; k4 < DHID / 4; ++k4) {
        const float4 av = arow[k4];
        const float4 xv = xrow[k4];
        const float4 w0l = ((const float4*)(Wld + 0 * DHID))[k4];
        const float4 w1l = ((const float4*)(Wld + 1 * DHID))[k4];
        const float4 w2l = ((const float4*)(Wld + 2 * DHID))[k4];
        const float4 w0r = ((const float4*)(Wrd + 0 * DHID))[k4];
        const float4 w1r = ((const float4*)(Wrd + 1 * DHID))[k4];
        const float4 w2r = ((const float4*)(Wrd + 2 * DHID))[k4];
        acc0 += (w0l.x * av.x + w0l.y * av.y + w0l.z * av.z + w0l.w * av.w) * 1.0f
              +  w0r.x * xv.x + w0r.y * xv.y + w0r.z * xv.z + w0r.w * xv.w
              + (w0l.x * 0.f); // keep shape; scale applied below
        acc1 +=  w1l.x * av.x + w1l.y * av.y + w1l.z * av.z + w1l.w * av.w
              +  w1r.x * xv.x + w1r.y * xv.y + w1r.z * xv.z + w1r.w * xv.w;
        acc2 +=  w2l.x * av.x + w2l.y * av.y + w2l.z * av.z + w2l.w * av.w
              +  w2r.x * xv.x + w2r.y * xv.y + w2r.z * xv.z + w2r.w * xv.w;
    }
    // NOTE: neighbor part must be scaled by invd; recompute split properly:
    // (done exactly below by a second pass over the neighbor term)
    float n0 = 0.f, n1 = 0.f, n2 = 0.f, s0 = 0.f, s1 = 0.f, s2 = 0.f;
    for (int k4 = 0; k4 < DHID / 4; ++k4) {
        const float4 av = arow[k4];
        const float4 xv = xrow[k4];
        const float4 w0l = ((const float4*)(Wld + 0 * DHID))[k4];
        const float4 w1l = ((const float4*)(Wld + 1 * DHID))[k4];
        const float4 w2l = ((const float4*)(Wld + 2 * DHID))[k4];
        const float4 w0r = ((const float4*)(Wrd + 0 * DHID))[k4];
        const float4 w1r = ((const float4*)(Wrd + 1 * DHID))[k4];
        const float4 w2r = ((const float4*)(Wrd + 2 * DHID))[k4];
        n0 += w0l.x * av.x + w0l.y * av.y + w0l.z * av.z + w0l.w * av.w;
        n1 += w1l.x * av.x + w1l.y * av.y + w1l.z * av.z + w1l.w * av.w;
        n2 += w2l.x * av.x + w2l.y * av.y + w2l.z * av.z + w2l.w * av.w;
        s0 += w0r.x * xv.x + w0r.y * xv.y + w0r.z * xv.z + w0r.w * xv.w;
        s1 += w1r.x * xv.x + w1r.y * xv.y + w1r.z * xv.z + w1r.w * xv.w;
        s2 += w2r.x * xv.x + w2r.y * xv.y + w2r.z * xv.z + w2r.w * xv.w;
    }
    (void)acc0; (void)acc1; (void)acc2;
    out[(size_t)row * 3 + 0] = n0 * invd + s0 + bld[0];
    out[(size_t)row * 3 + 1] = n1 * invd + s1 + bld[1];
    out[(size_t)row * 3 + 2] = n2 * invd + s2 + bld[2];
}

// ------------------------------------------------------------------ driver

extern "C" void kernel_launch(void* const* d_in, const int* in_sizes, int n_in,
                              void* d_out, int out_size, void* d_ws, size_t ws_size,
                              hipStream_t stream) {
    (void)in_sizes; (void)n_in; (void)out_size; (void)ws_size;

    const float* x   = (const float*)d_in[0];
    const int*   src = (const int*)  d_in[1];
    const int*   dst = (const int*)  d_in[2];
    const float* Wl1 = (const float*)d_in[3];
    const float* bl1 = (const float*)d_in[4];
    const float* Wr1 = (const float*)d_in[5];
    const float* Wl2 = (const float*)d_in[6];
    const float* bl2 = (const float*)d_in[7];
    const float* Wr2 = (const float*)d_in[8];
    const float* Wld = (const float*)d_in[9];
    const float* bld = (const float*)d_in[10];
    const float* Wrd = (const float*)d_in[11];

    float* out    = (float*)d_out;                    // (B,N,3) flat
    float* hidden = out + (size_t)BNROWS * 3;         // (B,N,128) flat

    // workspace carve-up
    float* deg    = (float*)d_ws;
    float* hbuf   = (float*)((char*)d_ws + (((size_t)NNODE * 4 + 255) & ~(size_t)255));
    float* aggbuf = hbuf + (size_t)BNROWS * DHID;     // 82 MB each

    const int ZB = 4096, THR = 256;
    const int ntiles  = BNROWS / 16;                  // 10000
    const int ublocks = (ntiles + 7) / 8;             // 8 waves / block
    const int eblocks = EDGES / 8;                    // 8 edges / block

    // degrees
    zero_f32v4<<<ZB, THR, 0, stream>>>((float4*)deg, (long)NNODE / 4);
    deg_kernel<<<(EDGES + THR - 1) / THR, THR, 0, stream>>>(dst, deg);

    // ---- layer 1: d_in = 64 -> h = relu(sage(x))
    zero_f32v4<<<ZB, THR, 0, stream>>>((float4*)aggbuf, (long)BNROWS * 64 / 4);
    scatter_add<6><<<eblocks, THR, 0, stream>>>(x, src, dst, aggbuf);
    sage_node_update<64, true, false><<<ublocks, THR, 0, stream>>>(
        aggbuf, x, Wl1, bl1, Wr1, deg, hbuf, nullptr, ntiles);

    // ---- layer 2: d_in = 128 -> hidden (raw, to d_out) + relu(hidden) in-place
    zero_f32v4<<<ZB, THR, 0, stream>>>((float4*)aggbuf, (long)BNROWS * 128 / 4);
    scatter_add<7><<<eblocks, THR, 0, stream>>>(hbuf, src, dst, aggbuf);
    sage_node_update<128, true, true><<<ublocks, THR, 0, stream>>>(
        aggbuf, hbuf, Wl2, bl2, Wr2, deg, hbuf, hidden, ntiles);

    // ---- layer 3: d_in = 128, d_out = 3 on relu(hidden) (= hbuf now)
    zero_f32v4<<<ZB, THR, 0, stream>>>((float4*)aggbuf, (long)BNROWS * 128 / 4);
    scatter_add<7><<<eblocks, THR, 0, stream>>>(hbuf, src, dst, aggbuf);
    sage_out_layer<<<(BNROWS + THR - 1) / THR, THR, 0, stream>>>(
        aggbuf, hbuf, Wld, bld, Wrd, deg, out);
}